// atrous_SS2D_23356032156047
// MI455X (gfx1250) — compile-verified
//
#include <hip/hip_runtime.h>
#include <math.h>

// Problem constants (from reference)
#define BSZ  2
#define HH   32
#define WW   32
#define LL   1024          // H*W
#define DM   256           // D_MODEL
#define DIN  512           // D_INNER
#define K4   4
#define RNK  16            // DT_RANK
#define NST  16            // D_STATE
#define C48  48            // DT_RANK + 2*D_STATE

typedef __attribute__((ext_vector_type(2))) float v2f;
typedef __attribute__((ext_vector_type(8))) float v8f;

// ---------------------------------------------------------------------------
// Register-blocked WMMA strip with COMPILE-TIME strides: one wave computes a
// 16 x (16*NB) output strip.  D_j = A(16xK) * W_j^T(Kx16), j = 0..NB-1, via
// V_WMMA_F32_16X16X4_F32.  With LDW/KD as constants, all B-loads are
// base + immediate-offset -> they clause together and overlap the 4
// independent WMMA accumulator chains instead of wait-serializing.
// ap/bp are already lane-adjusted: + (lane&15)*ld + 2*(lane>>4).
// ---------------------------------------------------------------------------
template <int NB, int LDW, int KD>
__device__ __forceinline__ void wmma_strip(const float* __restrict__ ap,
                                           const float* __restrict__ bp,
                                           v8f* acc) {
  const v8f zero = {0.f, 0.f, 0.f, 0.f, 0.f, 0.f, 0.f, 0.f};
#pragma unroll
  for (int j = 0; j < NB; ++j) acc[j] = zero;
  __builtin_prefetch(ap, 0, 3);                       // global_prefetch_b8
  __builtin_prefetch(bp, 0, 3);
#pragma unroll 4
  for (int k = 0; k < KD; k += 4) {
    v2f a = *(const v2f*)(ap + k);
#pragma unroll
    for (int j = 0; j < NB; ++j) {
      v2f b = *(const v2f*)(bp + j * 16 * LDW + k);   // constant offset
      acc[j] = __builtin_amdgcn_wmma_f32_16x16x4_f32(false, a, false, b,
                                                     (short)0, acc[j], false, false);
    }
  }
}

// ---------------------------------------------------------------------------
// Generic GEMM (compile-time shapes): C[M,N] = A[M,K] * W[N,K]^T.
// One 16x64 strip per wave (NB=4).  Grid launches exactly tilesM*groupsN
// waves (EXEC all ones for WMMA).
// ---------------------------------------------------------------------------
template <int LDA, int LDW, int LDC, int KD>
__global__ void gemm_wmma_kernel(const float* __restrict__ A, const float* __restrict__ Wt,
                                 float* __restrict__ C, int groupsN) {
  int strip = blockIdx.x * (blockDim.x >> 5) + (threadIdx.x >> 5);
  int tm = strip / groupsN, tg = strip % groupsN;
  const int lane = threadIdx.x & 31;
  const int rc   = lane & 15;          // A row / B col within tile
  const int koff = (lane >> 4) << 1;   // 0 or 2
  const float* ap = A  + (size_t)tm * 16 * LDA + rc * LDA + koff;
  const float* bp = Wt + (size_t)tg * 64 * LDW + rc * LDW + koff;
  v8f acc[4];
  wmma_strip<4, LDW, KD>(ap, bp, acc);
  const int rb = (lane >> 4) * 8;
  float* cp = C + (size_t)(tm * 16 + rb) * LDC + tg * 64 + rc;
#pragma unroll
  for (int j = 0; j < 4; ++j)
#pragma unroll
    for (int r = 0; r < 8; ++r) cp[r * LDC + j * 16] = acc[j][r];
}

// ---------------------------------------------------------------------------
// Depthwise 3x3 conv (pad 1) + bias + SiLU.  xz layout (b*L + l, 1024), first
// 512 channels are xx.  Output xc layout (b, d, h, w).
// ---------------------------------------------------------------------------
__global__ void conv_silu_kernel(const float* __restrict__ xz, const float* __restrict__ cw,
                                 const float* __restrict__ cb, float* __restrict__ xc) {
  int idx = blockIdx.x * blockDim.x + threadIdx.x;       // b<<19 | d<<10 | h<<5 | w
  int w = idx & 31, h = (idx >> 5) & 31, d = (idx >> 10) & 511, b = idx >> 19;
  float s = cb[d];
#pragma unroll
  for (int kh = 0; kh < 3; ++kh) {
    int hy = h + kh - 1;
    if (hy < 0 || hy >= HH) continue;
#pragma unroll
    for (int kw = 0; kw < 3; ++kw) {
      int wx = w + kw - 1;
      if (wx < 0 || wx >= WW) continue;
      s += xz[((size_t)(b * LL + hy * WW + wx)) * 1024 + d] * cw[d * 9 + kh * 3 + kw];
    }
  }
  xc[idx] = s / (1.f + __expf(-s));
}

// ---------------------------------------------------------------------------
// Build 4 scan orderings, transposed to (bk, l, d) for coalesced access.
// ---------------------------------------------------------------------------
__global__ void build_xs_kernel(const float* __restrict__ xc, float* __restrict__ xsT) {
  int idx = blockIdx.x * blockDim.x + threadIdx.x;       // bk<<19 | l<<9 | d
  int d = idx & 511, l = (idx >> 9) & 1023, bk = idx >> 19;
  int k = bk & 3, b = bk >> 2;
  int lp  = (k & 2) ? (LL - 1 - l) : l;                  // directions 2,3 reversed
  int pos = (k & 1) ? ((lp & 31) * 32 + (lp >> 5)) : lp; // directions 1,3 are WH order
  xsT[idx] = xc[(size_t)(b * DIN + d) * LL + pos];
}

// ---------------------------------------------------------------------------
// x_proj GEMM per (b,k): xdbl[bk][l][c] = sum_d xsT[bk][l][d] * xpw[k][c][d]
// M=1024, N=48 (exactly one NB=3 strip), K=512. 64 waves per bk.
// ---------------------------------------------------------------------------
__global__ void gemm_xproj_kernel(const float* __restrict__ xsT, const float* __restrict__ xpw,
                                  float* __restrict__ xdbl) {
  int bk = blockIdx.y, k = bk & 3;
  int tm = blockIdx.x * 8 + (threadIdx.x >> 5);
  const int lane = threadIdx.x & 31;
  const int rc   = lane & 15;
  const int koff = (lane >> 4) << 1;
  const float* ap = xsT + (size_t)bk * LL * DIN + (size_t)tm * 16 * DIN + rc * DIN + koff;
  const float* bp = xpw + (size_t)k * C48 * DIN + rc * DIN + koff;
  v8f acc[3];
  wmma_strip<3, DIN, DIN>(ap, bp, acc);
  const int rb = (lane >> 4) * 8;
  float* cp = xdbl + (size_t)bk * LL * C48 + (size_t)(tm * 16 + rb) * C48 + rc;
#pragma unroll
  for (int j = 0; j < 3; ++j)
#pragma unroll
    for (int r = 0; r < 8; ++r) cp[r * C48 + j * 16] = acc[j][r];
}

// ---------------------------------------------------------------------------
// dt projection per (b,k) + bias + softplus:
// delta[bk][l][d] = softplus( sum_r xdbl[bk][l][r] * dtw[k][d][r] + dtb[k][d] )
// M=1024, N=512, K=16.  NB=4 strips: 64 x 8 = 512 waves per bk.
// ---------------------------------------------------------------------------
__global__ void gemm_dtproj_kernel(const float* __restrict__ xdbl, const float* __restrict__ dtw,
                                   const float* __restrict__ dtb, float* __restrict__ delta) {
  int bk = blockIdx.y, k = bk & 3;
  int strip = blockIdx.x * 8 + (threadIdx.x >> 5);
  int tm = strip >> 3, tg = strip & 7;
  const int lane = threadIdx.x & 31;
  const int rc   = lane & 15;
  const int koff = (lane >> 4) << 1;
  const float* ap = xdbl + (size_t)bk * LL * C48 + (size_t)tm * 16 * C48 + rc * C48 + koff;
  const float* bp = dtw + (size_t)k * DIN * RNK + (size_t)tg * 64 * RNK + rc * RNK + koff;
  v8f acc[4];
  wmma_strip<4, RNK, RNK>(ap, bp, acc);
  const int rb = (lane >> 4) * 8;
  float* dp = delta + (size_t)bk * LL * DIN + (size_t)(tm * 16 + rb) * DIN + tg * 64 + rc;
#pragma unroll
  for (int j = 0; j < 4; ++j) {
    float bias = dtb[k * DIN + tg * 64 + j * 16 + rc];
#pragma unroll
    for (int r = 0; r < 8; ++r) {
      float v = acc[j][r] + bias;
      dp[(size_t)r * DIN + j * 16] = (v > 20.f) ? v : log1pf(__expf(v));
    }
  }
}

// ---------------------------------------------------------------------------
// Selective scan: one block per (b,k, d-chunk of 128). 16-state recurrence in
// registers, B/C broadcast through LDS each step. All global accesses to
// (bk,l,d)-major tensors are coalesced across d.
// ---------------------------------------------------------------------------
__global__ void scan_kernel(const float* __restrict__ xsT, const float* __restrict__ delta,
                            const float* __restrict__ xdbl, const float* __restrict__ A_logs,
                            const float* __restrict__ Ds, float* __restrict__ ys) {
  int bk = blockIdx.x >> 2;
  int k  = bk & 3;
  int d  = (blockIdx.x & 3) * 128 + threadIdx.x;
  int gd = k * DIN + d;                                  // row of A_logs / Ds
  float Arow[NST];
#pragma unroll
  for (int n = 0; n < NST; ++n) Arow[n] = -__expf(A_logs[gd * NST + n]);
  float Dv = Ds[gd];
  float h[NST];
#pragma unroll
  for (int n = 0; n < NST; ++n) h[n] = 0.f;

  __shared__ float sBC[32];
  const float* xd = xdbl  + (size_t)bk * LL * C48;
  const float* dl = delta + (size_t)bk * LL * DIN;
  const float* us = xsT   + (size_t)bk * LL * DIN;
  float*       yo = ys    + (size_t)bk * LL * DIN;

  for (int t = 0; t < LL; ++t) {
    if (threadIdx.x < 32) sBC[threadIdx.x] = xd[t * C48 + RNK + threadIdx.x];
    __syncthreads();
    float dt = dl[(size_t)t * DIN + d];
    float u  = us[(size_t)t * DIN + d];
    float du = dt * u;
    float y  = 0.f;
#pragma unroll
    for (int n = 0; n < NST; ++n) {
      h[n] = __expf(dt * Arow[n]) * h[n] + du * sBC[n];
      y += h[n] * sBC[16 + n];
    }
    yo[(size_t)t * DIN + d] = y + Dv * u;
    __syncthreads();
  }
}

// ---------------------------------------------------------------------------
// Merge 4 directions + LayerNorm(512) + SiLU(z) gate.  One block per token.
// ---------------------------------------------------------------------------
__global__ void fuse_ln_kernel(const float* __restrict__ ys, const float* __restrict__ xz,
                               const float* __restrict__ lnw, const float* __restrict__ lnb,
                               float* __restrict__ yg) {
  int b = blockIdx.x >> 10, l = blockIdx.x & 1023;
  int h = l >> 5, w = l & 31;
  int lwh = w * 32 + h;
  size_t b4 = (size_t)b * 4;
  const float* y0 = ys + ((b4 + 0) * LL + l) * DIN;
  const float* y1 = ys + ((b4 + 1) * LL + lwh) * DIN;
  const float* y2 = ys + ((b4 + 2) * LL + (LL - 1 - l)) * DIN;
  const float* y3 = ys + ((b4 + 3) * LL + (LL - 1 - lwh)) * DIN;

  __shared__ float red[256], red2[256];
  int tid = threadIdx.x, d2 = tid + 256;
  float v0 = y0[tid] + y1[tid] + y2[tid] + y3[tid];
  float v1 = y0[d2] + y1[d2] + y2[d2] + y3[d2];
  red[tid] = v0 + v1;
  red2[tid] = v0 * v0 + v1 * v1;
  __syncthreads();
  for (int s = 128; s > 0; s >>= 1) {
    if (tid < s) { red[tid] += red[tid + s]; red2[tid] += red2[tid + s]; }
    __syncthreads();
  }
  float mu  = red[0] * (1.f / 512.f);
  float var = red2[0] * (1.f / 512.f) - mu * mu;
  float rs  = rsqrtf(var + 1e-5f);

  const float* zrow = xz + ((size_t)b * LL + l) * 1024 + DIN;
  float* og = yg + ((size_t)b * LL + l) * DIN;
  {
    float yn = (v0 - mu) * rs * lnw[tid] + lnb[tid];
    float z = zrow[tid];
    og[tid] = yn * (z / (1.f + __expf(-z)));
  }
  {
    float yn = (v1 - mu) * rs * lnw[d2] + lnb[d2];
    float z = zrow[d2];
    og[d2] = yn * (z / (1.f + __expf(-z)));
  }
}

// ---------------------------------------------------------------------------
extern "C" void kernel_launch(void* const* d_in, const int* in_sizes, int n_in,
                              void* d_out, int out_size, void* d_ws, size_t ws_size,
                              hipStream_t stream) {
  (void)in_sizes; (void)n_in; (void)out_size; (void)ws_size;
  const float* x    = (const float*)d_in[0];
  const float* ipw  = (const float*)d_in[1];
  const float* cw   = (const float*)d_in[2];
  const float* cb   = (const float*)d_in[3];
  const float* xpw  = (const float*)d_in[4];
  const float* dtw  = (const float*)d_in[5];
  const float* dtb  = (const float*)d_in[6];
  const float* Alog = (const float*)d_in[7];
  const float* Dsp  = (const float*)d_in[8];
  const float* lnw  = (const float*)d_in[9];
  const float* lnb  = (const float*)d_in[10];
  const float* opw  = (const float*)d_in[11];
  float* out = (float*)d_out;

  // workspace carve (floats): total ~17.2M floats = ~68.7 MB
  float* ws    = (float*)d_ws;
  float* xz    = ws;                      // 2048*1024
  float* xc    = xz   + 2048 * 1024;      // 2*512*1024
  float* xsT   = xc   + 2 * 512 * 1024;   // 8*1024*512
  float* xdbl  = xsT  + 8 * 1024 * 512;   // 8*1024*48
  float* delta = xdbl + 8 * 1024 * 48;    // 8*1024*512
  float* ysc   = delta + 8 * 1024 * 512;  // 8*1024*512
  float* yg    = ysc  + 8 * 1024 * 512;   // 2048*512

  // 1) in_proj: xz[2048,1024] = x[2048,256] @ ipw[1024,256]^T
  //    128 x 16 strips (16x64 each) = 2048 waves
  gemm_wmma_kernel<DM, DM, 1024, DM><<<256, 256, 0, stream>>>(x, ipw, xz, 16);
  // 2) depthwise conv + SiLU
  conv_silu_kernel<<<4096, 256, 0, stream>>>(xz, cw, cb, xc);
  // 3) four scan orderings, (bk,l,d)-major
  build_xs_kernel<<<16384, 256, 0, stream>>>(xc, xsT);
  // 4) x_proj per (b,k): 64 NB=3 strips each
  gemm_xproj_kernel<<<dim3(8, 8), 256, 0, stream>>>(xsT, xpw, xdbl);
  // 5) dt proj + softplus per (b,k): 64 x 8 NB=4 strips each
  gemm_dtproj_kernel<<<dim3(64, 8), 256, 0, stream>>>(xdbl, dtw, dtb, delta);
  // 6) selective scan (latency-bound core)
  scan_kernel<<<32, 128, 0, stream>>>(xsT, delta, xdbl, Alog, Dsp, ysc);
  // 7) merge + LayerNorm + gate
  fuse_ln_kernel<<<2048, 256, 0, stream>>>(ysc, xz, lnw, lnb, yg);
  // 8) out_proj: out[2048,256] = yg[2048,512] @ opw[256,512]^T
  //    128 x 4 strips = 512 waves
  gemm_wmma_kernel<DIN, DIN, DM, DIN><<<64, 256, 0, stream>>>(yg, opw, out, 4);
}